// WeightedMSELossWithPenalty_28518582845592
// MI455X (gfx1250) — compile-verified
//
#include <hip/hip_runtime.h>

// CDNA5 / gfx1250, wave32.
typedef __attribute__((ext_vector_type(2))) float v2f;
typedef __attribute__((ext_vector_type(8))) float v8f;

// Wave32 sum via V_WMMA_F32_16X16X4_F32 (verified lowering in round 1):
// A (16x4 f32): lane L<16 -> M=L,{K0,K1}; lane L>=16 -> M=L-16,{K2,K3}.
// a = {p, 0}, B = ones -> D[m][n] = p_m + p_{m+16} for all n.
// Per-lane sum of the 8 C VGPRs gives rows 0-7 (lanes 0-15) / 8-15 (lanes 16-31);
// one shfl_xor(16) completes the wave total.
__device__ __forceinline__ float wave_sum_wmma(float p) {
    v2f a; a[0] = p;    a[1] = 0.0f;
    v2f b; b[0] = 1.0f; b[1] = 1.0f;
    v8f c = {};
    c = __builtin_amdgcn_wmma_f32_16x16x4_f32(
        false, a, false, b, (short)0, c, false, false);
    float s = c[0] + c[1] + c[2] + c[3] + c[4] + c[5] + c[6] + c[7];
    s += __shfl_xor(s, 16, 32);
    return s;  // full wave sum, uniform across lanes
}

__device__ __forceinline__ float range_pen(float v, float lo, float hi) {
    float b = v < lo ? (v - lo) : 0.0f;
    float a = v > hi ? (v - hi) : 0.0f;
    return b * b + a * a;
}

// Per-sample fused contribution.
// acc_w += sw*(0.5*(d0^2+d1^2) + d2^2); acc_p += range penalties on targets.
__device__ __forceinline__ void sample_contrib(
    float i0, float i1, float i2, int t0, int t1, int t2, int cs,
    const float* __restrict__ w_s, float& acc_w, float& acc_p)
{
    const float tf0 = (float)t0, tf1 = (float)t1, tf2 = (float)t2;
    const float sw  = w_s[t2 / 100 - 1];

    const float d0 = i0 - tf0, d1 = i1 - tf1, d2 = i2 - tf2;
    acc_w = fmaf(sw, fmaf(0.5f, fmaf(d0, d0, d1 * d1), d2 * d2), acc_w);

    // R02: lo = 11500 (idx==3 -> 6400, idx in 4..7 -> 6000), hi = 12000
    // R03: lo = 2200, hi = 2500 (idx==7 -> 13000)
    const int idx = cs - 1;
    float lo02 = 11500.0f;
    if (idx == 3)                  lo02 = 6400.0f;
    else if (idx >= 4 && idx <= 7) lo02 = 6000.0f;
    const float hi03 = (idx == 7) ? 13000.0f : 2500.0f;

    acc_p += range_pen(tf0, lo02, 12000.0f) + range_pen(tf1, 2200.0f, hi03);
}

__global__ __launch_bounds__(256) void wmse_partial_kernel(
    const float* __restrict__ inputs,       // [B,3] f32
    const int*   __restrict__ targets,      // [B,3] i32
    const int*   __restrict__ cycle_states, // [B]   i32
    const float* __restrict__ weights,      // [7]   f32
    int n, float invB,
    float* __restrict__ partials)           // [gridDim.x]
{
    __shared__ float w_s[8];
    __shared__ float wave_s[8];
    if (threadIdx.x < 7) w_s[threadIdx.x] = weights[threadIdx.x];
    __syncthreads();

    float acc_w = 0.0f;
    float acc_p = 0.0f;

    const float4* __restrict__ in4 = (const float4*)inputs;       // 3 per chunk
    const int4*   __restrict__ tg4 = (const int4*)targets;        // 3 per chunk
    const int4*   __restrict__ cs4 = (const int4*)cycle_states;   // 1 per chunk

    const int tid    = (int)(blockIdx.x * blockDim.x + threadIdx.x);
    const int stride = (int)(gridDim.x * blockDim.x);
    const int nc     = n >> 2;   // chunks of 4 samples; 16B-aligned b128 loads

    for (int c = tid; c < nc; c += stride) {
        if (c + stride < nc) {
            __builtin_prefetch(&in4[3 * (c + stride)], 0, 3);  // global_prefetch_b8
            __builtin_prefetch(&tg4[3 * (c + stride)], 0, 3);
        }
        const float4 a = in4[3 * c + 0];
        const float4 b = in4[3 * c + 1];
        const float4 d = in4[3 * c + 2];
        const int4   p = tg4[3 * c + 0];
        const int4   q = tg4[3 * c + 1];
        const int4   r = tg4[3 * c + 2];
        const int4   s = cs4[c];

        sample_contrib(a.x, a.y, a.z, p.x, p.y, p.z, s.x, w_s, acc_w, acc_p);
        sample_contrib(a.w, b.x, b.y, p.w, q.x, q.y, s.y, w_s, acc_w, acc_p);
        sample_contrib(b.z, b.w, d.x, q.z, q.w, r.x, s.z, w_s, acc_w, acc_p);
        sample_contrib(d.y, d.z, d.w, r.y, r.z, r.w, s.w, w_s, acc_w, acc_p);
    }

    // Scalar tail (empty when n % 4 == 0, e.g. B = 2^22).
    const int base = nc << 2;
    for (int i = base + tid; i < n; i += stride) {
        sample_contrib(inputs[3*i], inputs[3*i+1], inputs[3*i+2],
                       targets[3*i], targets[3*i+1], targets[3*i+2],
                       cycle_states[i], w_s, acc_w, acc_p);
    }

    // All threads converged: EXEC all-ones for WMMA.
    const float acc = acc_w * invB + acc_p;
    const float ws  = wave_sum_wmma(acc);

    const int wid = threadIdx.x >> 5;
    if ((threadIdx.x & 31) == 0) wave_s[wid] = ws;
    __syncthreads();
    if (threadIdx.x == 0) {
        float t = 0.0f;
        const int nw = blockDim.x >> 5;
        for (int k = 0; k < nw; ++k) t += wave_s[k];
        partials[blockIdx.x] = t;
    }
}

__global__ __launch_bounds__(256) void wmse_final_kernel(
    const float* __restrict__ partials, int np, float* __restrict__ out)
{
    __shared__ float wave_s[8];
    float acc = 0.0f;
    for (int i = threadIdx.x; i < np; i += blockDim.x) acc += partials[i];

    const float ws = wave_sum_wmma(acc);
    const int wid = threadIdx.x >> 5;
    if ((threadIdx.x & 31) == 0) wave_s[wid] = ws;
    __syncthreads();
    if (threadIdx.x == 0) {
        float t = 0.0f;
        const int nw = blockDim.x >> 5;
        for (int k = 0; k < nw; ++k) t += wave_s[k];
        out[0] = t;
    }
}

extern "C" void kernel_launch(void* const* d_in, const int* in_sizes, int n_in,
                              void* d_out, int out_size, void* d_ws, size_t ws_size,
                              hipStream_t stream) {
    const float* inputs       = (const float*)d_in[0];
    const int*   targets      = (const int*)d_in[1];
    const int*   cycle_states = (const int*)d_in[2];
    const float* weights      = (const float*)d_in[3];

    const int n = in_sizes[2];                 // B (cycle_states count)
    const float invB = 1.0f / (float)n;

    int nblk = 1024;
    if (ws_size < (size_t)nblk * sizeof(float)) {
        nblk = (int)(ws_size / sizeof(float));
        if (nblk < 1) nblk = 1;
    }
    float* partials = (float*)d_ws;

    wmse_partial_kernel<<<nblk, 256, 0, stream>>>(
        inputs, targets, cycle_states, weights, n, invB, partials);
    wmse_final_kernel<<<1, 256, 0, stream>>>(partials, nblk, (float*)d_out);
}